// AREMA_42047729828200
// MI455X (gfx1250) — compile-verified
//
#include <hip/hip_runtime.h>
#include <stdint.h>

// ---------------------------------------------------------------------------
// AREMA attack/release envelope follower for MI455X (gfx1250).
//
// x:   [B=8, T=16384, F=512] f32,  y_t = y_{t-1} + alpha*(x_t - y_{t-1})
//      alpha = A if (x_t - y_{t-1}) > 0 else R.
//
// Strategy: 4096 serial chains -> one chain per thread. Stream x through LDS
// with double-buffered GLOBAL_LOAD_ASYNC_TO_LDS_B128 (ASYNCcnt pipelining),
// which is the CDNA5-specific data-movement path for a latency-bound scan.
// ---------------------------------------------------------------------------

#define T_TOTAL 16384
#define F_TOTAL 512
#define TILE_T  32                       // timesteps per tile
#define FCHUNK  64                       // features (threads) per block
#define NTILES  (T_TOTAL / TILE_T)       // 512
#define TILE_BYTES (TILE_T * FCHUNK * 4) // 8192 B per buffer
#define NI      8                        // async b128 instrs per tile (4 rows each)

// Attack/release coefficients (1 - exp(-dt/tau)), dt=1ms, tau=5ms/50ms.
#define COEF_A 0.18126924692201815f
#define COEF_R 0.019801326693244656f

__device__ __forceinline__ void async_tile_load(const float* __restrict__ xbase,
                                                unsigned gthr_bytes,
                                                unsigned lds_bytes,
                                                unsigned tid) {
#pragma unroll
    for (int j = 0; j < NI; ++j) {
        // instr j covers rows 4j..4j+3 of the tile: 16 lanes x 16B per row.
        unsigned lds = lds_bytes + (unsigned)j * 1024u + tid * 16u;
        unsigned gof = gthr_bytes + (unsigned)j * (4u * F_TOTAL * 4u); // 4 rows * 2048B
        asm volatile("global_load_async_to_lds_b128 %0, %1, %2"
                     :: "v"(lds), "v"(gof), "s"(xbase)
                     : "memory");
    }
}

__global__ __launch_bounds__(FCHUNK)
void arema_scan_kernel(const float* __restrict__ x, float* __restrict__ out) {
    __shared__ float smem[2 * TILE_T * FCHUNK]; // 16 KB, double buffered

    const unsigned tid   = threadIdx.x;      // 0..63 -> feature within chunk
    const unsigned chunk = blockIdx.x;       // 0..7  -> feature chunk
    const unsigned bi    = blockIdx.y;       // 0..B-1 -> batch
    const unsigned f0    = chunk * FCHUNK;

    // DS-space byte address of the shared buffer: flat-aperture encoding keeps
    // the LDS offset in the low 32 bits of the generic pointer.
    const unsigned lds0 = (unsigned)(uintptr_t)(&smem[0]);

    // Per-thread global byte offset into x for the async lanes:
    // row within 4-row group = tid>>4, bytes within 256B row slice = (tid&15)*16.
    const unsigned base0 = bi * (unsigned)(T_TOTAL * F_TOTAL * 4) + f0 * 4u;
    const unsigned thr_g = base0 + (tid >> 4) * (unsigned)(F_TOTAL * 4) + (tid & 15u) * 16u;
    const unsigned tile_stride = (unsigned)(TILE_T * F_TOTAL * 4);  // 65536 B

    // Prime the double-buffer pipeline: tiles 0 and 1 in flight.
    async_tile_load(x, thr_g + 0u * tile_stride, lds0,               tid);
    async_tile_load(x, thr_g + 1u * tile_stride, lds0 + TILE_BYTES,  tid);

    float y = 0.0f;
    float* outp = out + (size_t)bi * T_TOTAL * F_TOTAL + f0 + tid;

    for (int k = 0; k < NTILES; ++k) {
        // Async loads complete in order: <=8 outstanding means everything
        // except the newest tile (k+1) has landed, i.e. buffer k is ready.
        if (k + 1 < NTILES) {
            asm volatile("s_wait_asynccnt 0x8" ::: "memory");
        } else {
            asm volatile("s_wait_asynccnt 0x0" ::: "memory");
        }
        __syncthreads();   // all waves' portions of buffer (k&1) visible

        const float* buf = smem + (size_t)(k & 1) * (TILE_T * FCHUNK);
#pragma unroll
        for (int tl = 0; tl < TILE_T; ++tl) {
            float xv = buf[tl * FCHUNK + tid];      // ds_load_b32, conflict-free
            float d  = xv - y;
            float a  = (d > 0.0f) ? COEF_A : COEF_R; // v_cmp + v_cndmask
            y = fmaf(a, d, y);                      // y + a*(x - y)
            outp[(size_t)(k * TILE_T + tl) * F_TOTAL] = y; // coalesced b32 store
        }

        __syncthreads();   // everyone done reading buffer (k&1)
        if (k + 2 < NTILES) {
            async_tile_load(x, thr_g + (unsigned)(k + 2) * tile_stride,
                            lds0 + (unsigned)(k & 1) * TILE_BYTES, tid);
        }
    }
}

extern "C" void kernel_launch(void* const* d_in, const int* in_sizes, int n_in,
                              void* d_out, int out_size, void* d_ws, size_t ws_size,
                              hipStream_t stream) {
    const float* x = (const float*)d_in[0];
    float* out = (float*)d_out;

    const int B = in_sizes[0] / (T_TOTAL * F_TOTAL);   // 8
    dim3 grid(F_TOTAL / FCHUNK, B);                    // 8 x 8 = 64 blocks
    dim3 block(FCHUNK);                                // 64 threads (2 waves)
    hipLaunchKernelGGL(arema_scan_kernel, grid, block, 0, stream, x, out);
}